// BioMedRelationExtractor_42700564857428
// MI455X (gfx1250) — compile-verified
//
#include <hip/hip_runtime.h>
#include <hip/hip_bf16.h>
#include <math.h>

typedef float v2f __attribute__((ext_vector_type(2)));
typedef float v8f __attribute__((ext_vector_type(8)));

static constexpr int B_ = 32, L_ = 300, D_ = 768, E_ = 600;
static constexpr int R_ = 26, H_ = 256, GO_ = 128;
static constexpr int HEADS_ = 8, HD_ = 32;
static constexpr int K_ = 9, CL_ = 150;
static constexpr int NPT_ = 32, PD_ = 8, NPC_ = NPT_ * CL_;   // 4800
static constexpr int OC_ = 5, OD_ = 16;

__device__ __forceinline__ v8f wmma4(v2f a, v2f b, v8f c) {
  return __builtin_amdgcn_wmma_f32_16x16x4_f32(false, a, false, b,
                                               (short)0, c, false, false);
}

// ---------------------------------------------------------------------------
// Edge messages: msg = x[b,src] @ W_rel[etype]  -> atomic scatter into agg[b,dst]
// ---------------------------------------------------------------------------
__global__ void edge_msg_kernel(const float* __restrict__ x,
                                const int* __restrict__ src,
                                const int* __restrict__ dst,
                                const int* __restrict__ et,
                                const float* __restrict__ W_rel,
                                float* __restrict__ agg) {
  int blk = blockIdx.x;
  int b = blk / E_;
  int e = blk - b * E_;
  int s = src[b * E_ + e];
  int d = dst[b * E_ + e];
  int r = et[b * E_ + e];
  __shared__ float xs[D_];
  int tid = threadIdx.x;
  for (int i = tid; i < D_; i += 256)
    xs[i] = x[((size_t)b * L_ + s) * D_ + i];
  __syncthreads();
  const float* W = W_rel + (size_t)r * D_ * H_ + tid;   // column tid
  float acc = 0.0f;
  for (int dd = 0; dd < D_; ++dd)
    acc += xs[dd] * W[(size_t)dd * H_];
  atomicAdd(&agg[((size_t)b * L_ + d) * H_ + tid], acc);
}

// ---------------------------------------------------------------------------
// f32 WMMA GEMM, one wave = 32x32 C tile (2x2 of 16x16), software-pipelined.
//   BT=true : B[k][n] = Bm[n*ldb + k]   (weight stored [N,K], float2 B loads)
//   BT=false: B[k][n] = Bm[k*ldb + n]   (weight stored [K,N], coalesced loads)
// Fragment layouts per CDNA5 ISA 7.12.2.
// ---------------------------------------------------------------------------
template <bool BT>
__global__ void gemm_wmma2_kernel(const float* __restrict__ A, int lda,
                                  const float* __restrict__ Bm, int ldb,
                                  float* __restrict__ C, int ldc,
                                  const float* __restrict__ bias, int addC,
                                  int M, int N, int Kdim) {
  int wave = (int)((blockIdx.x * blockDim.x + threadIdx.x) >> 5);
  int lane = threadIdx.x & 31;
  int tN = N >> 5;
  int nt = (M >> 5) * tN;
  if (wave >= nt) return;                 // whole-wave uniform exit
  int m0 = (wave / tN) << 5;
  int n0 = (wave % tN) << 5;
  int hi = lane >> 4;
  int l  = lane & 15;
  int col0 = n0 + l, col1 = n0 + 16 + l;

  v8f acc00, acc01, acc10, acc11;
#pragma unroll
  for (int v = 0; v < 8; ++v) {
    int r0 = m0 + v + 8 * hi, r1 = r0 + 16;
    if (addC) {
      acc00[v] = C[(size_t)r0 * ldc + col0];
      acc01[v] = C[(size_t)r0 * ldc + col1];
      acc10[v] = C[(size_t)r1 * ldc + col0];
      acc11[v] = C[(size_t)r1 * ldc + col1];
    } else {
      acc00[v] = acc01[v] = acc10[v] = acc11[v] = 0.0f;
    }
  }
  if (bias) {
    float bv0 = bias[col0], bv1 = bias[col1];
#pragma unroll
    for (int v = 0; v < 8; ++v) {
      acc00[v] += bv0; acc10[v] += bv0;
      acc01[v] += bv1; acc11[v] += bv1;
    }
  }

  const float* Ar0 = A + (size_t)(m0 + l) * lda;
  const float* Ar1 = A + (size_t)(m0 + 16 + l) * lda;

  auto loadB = [&](int ka, int col) -> v2f {
    if (BT) {
      return *(const v2f*)(Bm + (size_t)col * ldb + ka);
    } else {
      v2f r = { Bm[(size_t)ka * ldb + col], Bm[(size_t)(ka + 1) * ldb + col] };
      return r;
    }
  };

  int ka0 = 2 * hi;
  v2f a0 = *(const v2f*)(Ar0 + ka0);
  v2f a1 = *(const v2f*)(Ar1 + ka0);
  v2f b0 = loadB(ka0, col0);
  v2f b1 = loadB(ka0, col1);
  for (int kk = 4; kk < Kdim; kk += 4) {
    int ka = kk + 2 * hi;
    v2f na0 = *(const v2f*)(Ar0 + ka);
    v2f na1 = *(const v2f*)(Ar1 + ka);
    v2f nb0 = loadB(ka, col0);
    v2f nb1 = loadB(ka, col1);
    acc00 = wmma4(a0, b0, acc00);
    acc01 = wmma4(a0, b1, acc01);
    acc10 = wmma4(a1, b0, acc10);
    acc11 = wmma4(a1, b1, acc11);
    a0 = na0; a1 = na1; b0 = nb0; b1 = nb1;
  }
  acc00 = wmma4(a0, b0, acc00);
  acc01 = wmma4(a0, b1, acc01);
  acc10 = wmma4(a1, b0, acc10);
  acc11 = wmma4(a1, b1, acc11);

#pragma unroll
  for (int v = 0; v < 8; ++v) {
    int r0 = m0 + v + 8 * hi, r1 = r0 + 16;
    C[(size_t)r0 * ldc + col0] = acc00[v];
    C[(size_t)r0 * ldc + col1] = acc01[v];
    C[(size_t)r1 * ldc + col0] = acc10[v];
    C[(size_t)r1 * ldc + col1] = acc11[v];
  }
}

// ---------------------------------------------------------------------------
// Conv1d (stride 2, pad 4) as one of 9 shifted GEMMs (koff = 0..8):
//   C[(b,o), oc] += sum_d X[b, 2o-4+koff, d] * conv_w[oc, d, koff]
// One wave = 32x32 C tile.  M = B*CL = 4800, N = 256, K = 768.
// ---------------------------------------------------------------------------
__global__ void conv_gemm_wmma_kernel(const float* __restrict__ X,
                                      const float* __restrict__ Wc,
                                      float* __restrict__ C,
                                      const float* __restrict__ bias,
                                      int koff, int addC) {
  int wave = (int)((blockIdx.x * blockDim.x + threadIdx.x) >> 5);
  int lane = threadIdx.x & 31;
  const int NN = NPT_ * PD_;                  // 256
  const int tN = NN >> 5;                     // 8
  const int nt = ((B_ * CL_) >> 5) * tN;      // 1200
  if (wave >= nt) return;
  int m0 = (wave / tN) << 5;
  int n0 = (wave % tN) << 5;
  int hi = lane >> 4;
  int l  = lane & 15;
  int col0 = n0 + l, col1 = n0 + 16 + l;

  auto rowPtr = [&](int row, float& mask) -> const float* {
    int b = row / CL_;
    int o = row - b * CL_;
    int in = 2 * o - 4 + koff;
    bool valid = (in >= 0) && (in < L_);
    mask = valid ? 1.0f : 0.0f;
    int inc = valid ? in : 0;
    return X + ((size_t)b * L_ + inc) * D_;
  };
  float mask0, mask1;
  const float* Ar0 = rowPtr(m0 + l, mask0);
  const float* Ar1 = rowPtr(m0 + 16 + l, mask1);

  v8f acc00, acc01, acc10, acc11;
#pragma unroll
  for (int v = 0; v < 8; ++v) {
    int r0 = m0 + v + 8 * hi, r1 = r0 + 16;
    if (addC) {
      acc00[v] = C[(size_t)r0 * NN + col0];
      acc01[v] = C[(size_t)r0 * NN + col1];
      acc10[v] = C[(size_t)r1 * NN + col0];
      acc11[v] = C[(size_t)r1 * NN + col1];
    } else {
      acc00[v] = acc01[v] = acc10[v] = acc11[v] = 0.0f;
    }
  }
  if (bias) {
    float bv0 = bias[col0], bv1 = bias[col1];
#pragma unroll
    for (int v = 0; v < 8; ++v) {
      acc00[v] += bv0; acc10[v] += bv0;
      acc01[v] += bv1; acc11[v] += bv1;
    }
  }

  const float* Wb0 = Wc + (size_t)col0 * (D_ * K_) + koff;
  const float* Wb1 = Wc + (size_t)col1 * (D_ * K_) + koff;
  auto loadW = [&](const float* Wb, int ka) -> v2f {
    v2f r = { Wb[(size_t)ka * K_], Wb[(size_t)(ka + 1) * K_] };
    return r;
  };
  auto loadA = [&](const float* Ar, float mask, int ka) -> v2f {
    v2f r = *(const v2f*)(Ar + ka);
    r.x *= mask; r.y *= mask;
    return r;
  };

  int ka0 = 2 * hi;
  v2f a0 = loadA(Ar0, mask0, ka0);
  v2f a1 = loadA(Ar1, mask1, ka0);
  v2f b0 = loadW(Wb0, ka0);
  v2f b1 = loadW(Wb1, ka0);
  for (int kk = 4; kk < D_; kk += 4) {
    int ka = kk + 2 * hi;
    v2f na0 = loadA(Ar0, mask0, ka);
    v2f na1 = loadA(Ar1, mask1, ka);
    v2f nb0 = loadW(Wb0, ka);
    v2f nb1 = loadW(Wb1, ka);
    acc00 = wmma4(a0, b0, acc00);
    acc01 = wmma4(a0, b1, acc01);
    acc10 = wmma4(a1, b0, acc10);
    acc11 = wmma4(a1, b1, acc11);
    a0 = na0; a1 = na1; b0 = nb0; b1 = nb1;
  }
  acc00 = wmma4(a0, b0, acc00);
  acc01 = wmma4(a0, b1, acc01);
  acc10 = wmma4(a1, b0, acc10);
  acc11 = wmma4(a1, b1, acc11);

#pragma unroll
  for (int v = 0; v < 8; ++v) {
    int r0 = m0 + v + 8 * hi, r1 = r0 + 16;
    C[(size_t)r0 * NN + col0] = acc00[v];
    C[(size_t)r0 * NN + col1] = acc01[v];
    C[(size_t)r1 * NN + col0] = acc10[v];
    C[(size_t)r1 * NN + col1] = acc11[v];
  }
}

// ---------------------------------------------------------------------------
// Streaming attention: one block = one (b, head, query row n). 64 threads.
// ---------------------------------------------------------------------------
__global__ void attn_kernel(const float* __restrict__ qkv,
                            float* __restrict__ out) {
  int blk = blockIdx.x;                 // B*HEADS*L
  int n = blk % L_;
  int t = blk / L_;
  int h = t % HEADS_;
  int b = t / HEADS_;
  int tid = threadIdx.x;                // 64

  __shared__ float qs[HD_];
  __shared__ float sc[L_];
  __shared__ float red[64];
  __shared__ float sres;

  const float* base = qkv + (size_t)b * L_ * (3 * H_);
  if (tid < HD_) qs[tid] = base[(size_t)n * (3 * H_) + h * HD_ + tid];
  __syncthreads();

  float lmax = -1e30f;
  for (int m = tid; m < L_; m += 64) {
    const float* kr = base + (size_t)m * (3 * H_) + H_ + h * HD_;
    float s = 0.0f;
#pragma unroll
    for (int d2 = 0; d2 < HD_; ++d2) s += qs[d2] * kr[d2];
    s *= 0.17677669529663687f;          // 1/sqrt(32)
    sc[m] = s;
    lmax = fmaxf(lmax, s);
  }
  red[tid] = lmax;
  __syncthreads();
  for (int s = 32; s > 0; s >>= 1) {
    if (tid < s) red[tid] = fmaxf(red[tid], red[tid + s]);
    __syncthreads();
  }
  if (tid == 0) sres = red[0];
  __syncthreads();
  float mx = sres;
  __syncthreads();

  float lsum = 0.0f;
  for (int m = tid; m < L_; m += 64) {
    float e = expf(sc[m] - mx);
    sc[m] = e;
    lsum += e;
  }
  red[tid] = lsum;
  __syncthreads();
  for (int s = 32; s > 0; s >>= 1) {
    if (tid < s) red[tid] += red[tid + s];
    __syncthreads();
  }
  if (tid == 0) sres = red[0];
  __syncthreads();
  float inv = 1.0f / sres;

  if (tid < HD_) {
    float acc = 0.0f;
    for (int m = 0; m < L_; ++m)
      acc += sc[m] * base[(size_t)m * (3 * H_) + 2 * H_ + h * HD_ + tid];
    out[((size_t)b * L_ + n) * H_ + h * HD_ + tid] = acc * inv;
  }
}

// mean over L
__global__ void pool_kernel(const float* __restrict__ h, float* __restrict__ pooled) {
  int b = blockIdx.x, j = threadIdx.x;
  float acc = 0.0f;
  for (int n = 0; n < L_; ++n) acc += h[((size_t)b * L_ + n) * H_ + j];
  pooled[b * H_ + j] = acc * (1.0f / (float)L_);
}

// prim repack + squash over PD
__global__ void squash_prim_kernel(const float* __restrict__ convC,
                                   float* __restrict__ u) {
  int idx = blockIdx.x * 256 + threadIdx.x;     // B*NPC
  if (idx >= B_ * NPC_) return;
  int b = idx / NPC_;
  int i = idx - b * NPC_;
  int t = i / CL_;
  int o = i - t * CL_;
  const float* src = convC + ((size_t)b * CL_ + o) * (NPT_ * PD_) + t * PD_;
  float v[PD_];
  float sn = 0.0f;
#pragma unroll
  for (int p = 0; p < PD_; ++p) { v[p] = src[p]; sn += v[p] * v[p]; }
  float scale = (sn / (1.0f + sn)) / (sqrtf(sn + 1e-8f) + 1e-8f);
  float* dst = u + (size_t)idx * PD_;
#pragma unroll
  for (int p = 0; p < PD_; ++p) dst[p] = v[p] * scale;
}

// u_hat[b,i,oc,od] = sum_pd caps_W[i,oc,od,pd] * u[b,i,pd]
__global__ void uhat_kernel(const float* __restrict__ caps_W,
                            const float* __restrict__ u,
                            float* __restrict__ uhat) {
  int i = blockIdx.x;                  // NPC
  __shared__ float Ws[OC_ * OD_ * PD_];// 640
  int tid = threadIdx.x;               // 256
  for (int j = tid; j < OC_ * OD_ * PD_; j += 256)
    Ws[j] = caps_W[(size_t)i * (OC_ * OD_ * PD_) + j];
  __syncthreads();
  for (int idx = tid; idx < B_ * OC_ * OD_; idx += 256) {
    int b = idx / (OC_ * OD_);
    int op = idx - b * (OC_ * OD_);
    const float* uv = u + ((size_t)b * NPC_ + i) * PD_;
    const float* w = Ws + op * PD_;
    float acc = 0.0f;
#pragma unroll
    for (int p = 0; p < PD_; ++p) acc += w[p] * uv[p];
    uhat[((size_t)b * NPC_ + i) * (OC_ * OD_) + op] = acc;
  }
}

// s_j = sum_i c * u_hat ; v_j = squash(s_j). One block per (b, oc).
__global__ void route_s_kernel(const float* __restrict__ uhat,
                               const float* __restrict__ bij,
                               float* __restrict__ vj, int uniform) {
  int blk = blockIdx.x;
  int b = blk / OC_;
  int oc = blk - b * OC_;
  int tid = threadIdx.x;               // 256
  float acc[OD_];
#pragma unroll
  for (int j = 0; j < OD_; ++j) acc[j] = 0.0f;
  for (int i = tid; i < NPC_; i += 256) {
    float c;
    if (uniform) {
      c = 1.0f / (float)OC_;
    } else {
      const float* br = bij + ((size_t)b * NPC_ + i) * OC_;
      float m = br[0];
#pragma unroll
      for (int j = 1; j < OC_; ++j) m = fmaxf(m, br[j]);
      float den = 0.0f, ex = 0.0f;
#pragma unroll
      for (int j = 0; j < OC_; ++j) {
        float e = expf(br[j] - m);
        den += e;
        if (j == oc) ex = e;
      }
      c = ex / den;
    }
    const float* uh = uhat + (((size_t)b * NPC_ + i) * OC_ + oc) * OD_;
#pragma unroll
    for (int j = 0; j < OD_; ++j) acc[j] += c * uh[j];
  }
  __shared__ float red[256];
  __shared__ float sj[OD_];
  for (int j = 0; j < OD_; ++j) {
    red[tid] = acc[j];
    __syncthreads();
    for (int s = 128; s > 0; s >>= 1) {
      if (tid < s) red[tid] += red[tid + s];
      __syncthreads();
    }
    if (tid == 0) sj[j] = red[0];
    __syncthreads();
  }
  if (tid == 0) {
    float sn = 0.0f;
#pragma unroll
    for (int j = 0; j < OD_; ++j) sn += sj[j] * sj[j];
    float scale = (sn / (1.0f + sn)) / (sqrtf(sn + 1e-8f) + 1e-8f);
#pragma unroll
    for (int j = 0; j < OD_; ++j)
      vj[((size_t)b * OC_ + oc) * OD_ + j] = sj[j] * scale;
  }
}

// b_ij (+)= <u_hat[b,i,oc,:], v_j[b,oc,:]>
__global__ void update_b_kernel(const float* __restrict__ uhat,
                                const float* __restrict__ vj,
                                float* __restrict__ bij, int first) {
  int idx = blockIdx.x * 256 + threadIdx.x;     // B*NPC*OC
  if (idx >= B_ * NPC_ * OC_) return;
  int oc = idx % OC_;
  int bi = idx / OC_;
  int b = bi / NPC_;
  const float* uh = uhat + (size_t)idx * OD_;
  const float* vv = vj + ((size_t)b * OC_ + oc) * OD_;
  float d = 0.0f;
#pragma unroll
  for (int j = 0; j < OD_; ++j) d += uh[j] * vv[j];
  if (first) bij[idx] = d; else bij[idx] += d;
}

// final FC over concat(gcn_out[128], caps_out[80])
__global__ void fc_kernel(const float* __restrict__ gcn_out,
                          const float* __restrict__ vj,
                          const float* __restrict__ fc_w,
                          const float* __restrict__ fc_b,
                          float* __restrict__ out) {
  int idx = threadIdx.x;
  if (idx >= B_ * OC_) return;
  int b = idx / OC_;
  int oc = idx - b * OC_;
  const float* w = fc_w + oc * (GO_ + OC_ * OD_);
  float acc = fc_b[oc];
  for (int f = 0; f < GO_; ++f) acc += gcn_out[b * GO_ + f] * w[f];
  for (int f = 0; f < OC_ * OD_; ++f) acc += vj[b * OC_ * OD_ + f] * w[GO_ + f];
  out[idx] = acc;
}

// ---------------------------------------------------------------------------
extern "C" void kernel_launch(void* const* d_in, const int* in_sizes, int n_in,
                              void* d_out, int out_size, void* d_ws, size_t ws_size,
                              hipStream_t stream) {
  const float* x        = (const float*)d_in[0];
  const int*   g_src    = (const int*)d_in[1];
  const int*   g_dst    = (const int*)d_in[2];
  const int*   g_et     = (const int*)d_in[3];
  const float* W_rel    = (const float*)d_in[4];
  const float* loop_w   = (const float*)d_in[5];
  const float* gcn_b    = (const float*)d_in[6];
  const float* in_w     = (const float*)d_in[7];
  const float* in_b     = (const float*)d_in[8];
  const float* out_w    = (const float*)d_in[9];
  const float* out_b    = (const float*)d_in[10];
  const float* mlp_w    = (const float*)d_in[11];
  const float* mlp_b    = (const float*)d_in[12];
  const float* conv_w   = (const float*)d_in[13];
  const float* conv_b   = (const float*)d_in[14];
  const float* caps_W   = (const float*)d_in[15];
  const float* fc_w     = (const float*)d_in[16];
  const float* fc_b     = (const float*)d_in[17];
  float* out = (float*)d_out;

  float* ws = (float*)d_ws;
  size_t off = 0;
  float* agg    = ws + off; off += (size_t)B_ * L_ * H_;          // h, then h2
  float* qkv    = ws + off; off += (size_t)B_ * L_ * 3 * H_;
  float* attn   = ws + off; off += (size_t)B_ * L_ * H_;
  float* pooled = ws + off; off += (size_t)B_ * H_;
  float* gcnout = ws + off; off += (size_t)B_ * GO_;
  float* convC  = ws + off; off += (size_t)B_ * CL_ * NPT_ * PD_;
  float* u      = ws + off; off += (size_t)B_ * NPC_ * PD_;
  float* uhat   = ws + off; off += (size_t)B_ * NPC_ * OC_ * OD_;
  float* bij    = ws + off; off += (size_t)B_ * NPC_ * OC_;
  float* vj     = ws + off; off += (size_t)B_ * OC_ * OD_;
  float* h2 = agg;   // reuse: h dead after QKV GEMM

  // ---- GCN branch
  hipMemsetAsync(agg, 0, (size_t)B_ * L_ * H_ * sizeof(float), stream);
  edge_msg_kernel<<<B_ * E_, 256, 0, stream>>>(x, g_src, g_dst, g_et, W_rel, agg);

  // h = agg + x @ loop_w + gcn_b   (M=9600,N=256,K=768; B normal [K,N])
  {
    int tiles = (B_ * L_ / 32) * (H_ / 32);
    gemm_wmma2_kernel<false><<<(tiles + 7) / 8, 256, 0, stream>>>(
        x, D_, loop_w, H_, agg, H_, gcn_b, 1, B_ * L_, H_, D_);
  }
  // qkv = h @ in_proj_w^T + in_b   (M=9600,N=768,K=256; B stored [N,K])
  {
    int tiles = (B_ * L_ / 32) * (3 * H_ / 32);
    gemm_wmma2_kernel<true><<<(tiles + 7) / 8, 256, 0, stream>>>(
        agg, H_, in_w, H_, qkv, 3 * H_, in_b, 0, B_ * L_, 3 * H_, H_);
  }
  attn_kernel<<<B_ * HEADS_ * L_, 64, 0, stream>>>(qkv, attn);
  // h2 = attn @ out_proj_w^T + out_b
  {
    int tiles = (B_ * L_ / 32) * (H_ / 32);
    gemm_wmma2_kernel<true><<<(tiles + 7) / 8, 256, 0, stream>>>(
        attn, H_, out_w, H_, h2, H_, out_b, 0, B_ * L_, H_, H_);
  }
  pool_kernel<<<B_, H_, 0, stream>>>(h2, pooled);
  // gcn_out = pooled @ mlp_w^T + mlp_b   (M=32,N=128,K=256)
  {
    int tiles = (B_ / 32) * (GO_ / 32);
    gemm_wmma2_kernel<true><<<(tiles + 7) / 8, 256, 0, stream>>>(
        pooled, H_, mlp_w, H_, gcnout, GO_, mlp_b, 0, B_, GO_, H_);
  }

  // ---- Capsule branch: conv1d as 9 accumulating GEMMs
  {
    int tiles = (B_ * CL_ / 32) * ((NPT_ * PD_) / 32);  // 1200 waves
    for (int k = 0; k < K_; ++k) {
      conv_gemm_wmma_kernel<<<(tiles + 7) / 8, 256, 0, stream>>>(
          x, conv_w, convC, (k == 0) ? conv_b : nullptr, k, (k == 0) ? 0 : 1);
    }
  }
  squash_prim_kernel<<<(B_ * NPC_ + 255) / 256, 256, 0, stream>>>(convC, u);
  uhat_kernel<<<NPC_, 256, 0, stream>>>(caps_W, u, uhat);

  // Dynamic routing, 3 iterations
  route_s_kernel<<<B_ * OC_, 256, 0, stream>>>(uhat, bij, vj, 1);
  update_b_kernel<<<(B_ * NPC_ * OC_ + 255) / 256, 256, 0, stream>>>(uhat, vj, bij, 1);
  route_s_kernel<<<B_ * OC_, 256, 0, stream>>>(uhat, bij, vj, 0);
  update_b_kernel<<<(B_ * NPC_ * OC_ + 255) / 256, 256, 0, stream>>>(uhat, vj, bij, 0);
  route_s_kernel<<<B_ * OC_, 256, 0, stream>>>(uhat, bij, vj, 0);

  // ---- Head
  fc_kernel<<<1, 256, 0, stream>>>(gcnout, vj, fc_w, fc_b, out);
}